// CSG_layer_89790586290259
// MI455X (gfx1250) — compile-verified
//
#include <hip/hip_runtime.h>
#include <math.h>

typedef __attribute__((ext_vector_type(2))) float v2f;
typedef __attribute__((ext_vector_type(8))) float v8f;

#define EPSF 1.1920928955078125e-7f   // FLT_EPSILON

constexpr int B      = 32;
constexpr int P      = 8192;
constexpr int S_IN   = 64;
constexpr int S_OUT  = 32;
constexpr int L      = 256;
constexpr int ENC_IN = S_IN * 2 * S_OUT;   // 4096
constexpr int Y_COLS = 4 * S_OUT;          // 128
constexpr int ROWS_PER_BLOCK = 128;        // 8 waves x 16 rows

__device__ __forceinline__ float clip01(float v) {
    return fminf(fmaxf(v, 0.0f), 1.0f);
}

// ---------------------------------------------------------------------------
// Kernel 1: per-batch control math (logits -> softmax -> encoder -> gumbel
// masks). Masks are written to workspace in WMMA-B fragment order:
//   frag f = kc*2 + nc   (kc = K-chunk of 4 rows, nc = N-chunk of 16 cols)
//   within frag: lane L holds (K = 4kc + 2*(L>>4) + r, N = 16nc + (L&15))
//   at offset f*64 + L*2 + r  -> one ds_load_b64 per B fragment in kernel 2.
// ---------------------------------------------------------------------------
__global__ __launch_bounds__(256) void csg_small_kernel(
    const float* __restrict__ latent, const float* __restrict__ gu,
    const float* __restrict__ Kl,     const float* __restrict__ Kr,
    const float* __restrict__ temp,   const float* __restrict__ W1,
    const float* __restrict__ b1,     const float* __restrict__ W2,
    const float* __restrict__ b2,     float* __restrict__ mask_ws,
    float* __restrict__ venc)
{
    __shared__ float lat[L];
    __shared__ float V[2 * 2048];   // [side][i*32+o]
    __shared__ float h[L];

    const int t = threadIdx.x;
    const int b = blockIdx.x;

    lat[t] = latent[b * L + t];
    __syncthreads();

    // ---- logits: V[side][i,o] = sum_l lat[l] * K_side[l, i, o] ----
    for (int rep = 0; rep < 16; ++rep) {
        int flat = rep * 256 + t;
        int side = flat >> 11;
        int io   = flat & 2047;
        const float* __restrict__ Kp = side ? Kr : Kl;
        float acc = 0.0f;
        #pragma unroll 8
        for (int l = 0; l < L; ++l)
            acc = fmaf(lat[l], Kp[l * 2048 + io], acc);
        V[flat] = acc;
    }
    __syncthreads();

    // ---- softmax over i (axis=-2) per (side, o) column ----
    if (t < 64) {
        int side = t >> 5, o = t & 31;
        float* col = &V[side * 2048 + o];
        float m = -INFINITY;
        for (int i = 0; i < S_IN; ++i) m = fmaxf(m, col[i * 32]);
        float s = 0.0f;
        for (int i = 0; i < S_IN; ++i) {
            float e = __expf(col[i * 32] - m);
            col[i * 32] = e;
            s += e;
        }
        float inv = 1.0f / s;
        for (int i = 0; i < S_IN; ++i) col[i * 32] *= inv;
    }
    __syncthreads();

    // ---- encoder: h = leaky(V_flat . W1 + b1); venc = h . W2 + b2 ----
    {
        float acc = b1[t];
        for (int k = 0; k < ENC_IN; ++k)
            acc = fmaf(V[k], W1[k * L + t], acc);
        h[t] = (acc >= 0.0f) ? acc : 0.01f * acc;
    }
    __syncthreads();
    {
        float acc = b2[t];
        for (int k = 0; k < L; ++k)
            acc = fmaf(h[k], W2[k * L + t], acc);
        venc[b * L + t] = acc;
    }

    // ---- gumbel-softmax masks -> fragment-ordered workspace ----
    if (t < 64) {
        int side = t >> 5, o = t & 31;
        float tc  = fminf(fmaxf(temp[0], EPSF), 2.0f);
        float itc = 1.0f / tc;
        float* col = &V[side * 2048 + o];
        const float* __restrict__ gcol = gu + ((size_t)(b * 2 + side) * S_IN) * S_OUT + o;

        float m = -INFINITY;
        for (int i = 0; i < S_IN; ++i) {
            float u = fmaxf(gcol[i * S_OUT], EPSF);
            float g = -__logf(-__logf(u) + EPSF);
            float s = (__logf(col[i * 32] + EPSF) + g) * itc;
            col[i * 32] = s;
            m = fmaxf(m, s);
        }
        float ssum = 0.0f;
        for (int i = 0; i < S_IN; ++i) {
            float e = __expf(col[i * 32] - m);
            col[i * 32] = e;
            ssum += e;
        }
        float inv = 1.0f / ssum;

        float* __restrict__ wsb = mask_ws + (size_t)(b * 2 + side) * 2048;
        for (int i = 0; i < S_IN; ++i) {
            float mval = col[i * 32] * inv;
            int kc = i >> 2;
            int hi = (i >> 1) & 1;
            int r  = i & 1;
            int nc = o >> 4;
            int lane = hi * 16 + (o & 15);
            wsb[(kc * 2 + nc) * 64 + lane * 2 + r] = mval;
        }
    }
}

// ---------------------------------------------------------------------------
// Kernel 2: bandwidth-bound batched GEMM + epilogue.
//   yl/yr[b] = x[b](8192x64) . mask_{l,r}[b](64x32)   via V_WMMA_F32_16X16X4_F32
//   y = clip01(concat(yl+yr, yl+yr-1, yl-yr, yr-yl))
// One wave = 16-row slab; 16 K-steps x 4 WMMAs; masks staged in LDS (16 KB).
// ---------------------------------------------------------------------------
__global__ __launch_bounds__(256) void csg_gemm_kernel(
    const float* __restrict__ x, const float* __restrict__ mask_ws,
    float* __restrict__ y)
{
    __shared__ __align__(16) float fl[2048];
    __shared__ __align__(16) float fr[2048];

    const int t = threadIdx.x;
    const int blocksPerB = P / ROWS_PER_BLOCK;           // 64
    const int b      = blockIdx.x / blocksPerB;
    const int rowBlk = blockIdx.x % blocksPerB;

    const float* __restrict__ wl = mask_ws + (size_t)(b * 2 + 0) * 2048;
    const float* __restrict__ wr = mask_ws + (size_t)(b * 2 + 1) * 2048;
    #pragma unroll
    for (int rep = 0; rep < 8; ++rep) {
        int idx = rep * 256 + t;
        fl[idx] = wl[idx];
        fr[idx] = wr[idx];
    }
    __syncthreads();

    const int wave  = t >> 5;
    const int lane  = t & 31;
    const int mlane = lane & 15;
    const int hi    = lane >> 4;
    const int rowBase = rowBlk * ROWS_PER_BLOCK + wave * 16;

    // A-matrix 16x4 f32 layout: lane L -> row M = L&15;
    // VGPR0/1 hold K = 2*(L>>4) + {0,1}. Per K-chunk this is one float2 load.
    const float* __restrict__ xrow =
        x + ((size_t)b * P + rowBase + mlane) * S_IN + 2 * hi;

    v2f av[16];
    #pragma unroll
    for (int kc = 0; kc < 16; ++kc)
        av[kc] = *reinterpret_cast<const v2f*>(xrow + 4 * kc);

    v8f aL0 = {}, aL1 = {}, aR0 = {}, aR1 = {};
    #pragma unroll
    for (int kc = 0; kc < 16; ++kc) {
        v2f a = av[kc];
        v2f bl0 = *reinterpret_cast<const v2f*>(&fl[(kc * 2 + 0) * 64 + lane * 2]);
        v2f bl1 = *reinterpret_cast<const v2f*>(&fl[(kc * 2 + 1) * 64 + lane * 2]);
        v2f br0 = *reinterpret_cast<const v2f*>(&fr[(kc * 2 + 0) * 64 + lane * 2]);
        v2f br1 = *reinterpret_cast<const v2f*>(&fr[(kc * 2 + 1) * 64 + lane * 2]);
        aL0 = __builtin_amdgcn_wmma_f32_16x16x4_f32(false, a, false, bl0, (short)0, aL0, false, false);
        aL1 = __builtin_amdgcn_wmma_f32_16x16x4_f32(false, a, false, bl1, (short)0, aL1, false, false);
        aR0 = __builtin_amdgcn_wmma_f32_16x16x4_f32(false, a, false, br0, (short)0, aR0, false, false);
        aR1 = __builtin_amdgcn_wmma_f32_16x16x4_f32(false, a, false, br1, (short)0, aR1, false, false);
    }

    // C/D 16x16 f32 layout: lane L -> N = L&15, VGPR v -> M = v + 8*(L>>4).
    float* __restrict__ yb = y + (size_t)b * P * Y_COLS;
    #pragma unroll
    for (int v = 0; v < 8; ++v) {
        int row = rowBase + v + 8 * hi;
        float* __restrict__ yrow = yb + (size_t)row * Y_COLS;
        {
            float l = aL0[v], r = aR0[v];
            int col = mlane;
            float s = l + r;
            yrow[col]      = clip01(s);
            yrow[32 + col] = clip01(s - 1.0f);
            yrow[64 + col] = clip01(l - r);
            yrow[96 + col] = clip01(r - l);
        }
        {
            float l = aL1[v], r = aR1[v];
            int col = 16 + mlane;
            float s = l + r;
            yrow[col]      = clip01(s);
            yrow[32 + col] = clip01(s - 1.0f);
            yrow[64 + col] = clip01(l - r);
            yrow[96 + col] = clip01(r - l);
        }
    }
}

extern "C" void kernel_launch(void* const* d_in, const int* in_sizes, int n_in,
                              void* d_out, int out_size, void* d_ws, size_t ws_size,
                              hipStream_t stream) {
    const float* x      = (const float*)d_in[0];
    const float* latent = (const float*)d_in[1];
    const float* gu     = (const float*)d_in[2];
    const float* Kl     = (const float*)d_in[3];
    const float* Kr     = (const float*)d_in[4];
    const float* temp   = (const float*)d_in[5];
    const float* W1     = (const float*)d_in[6];
    const float* b1     = (const float*)d_in[7];
    const float* W2     = (const float*)d_in[8];
    const float* b2     = (const float*)d_in[9];

    float* y    = (float*)d_out;                        // (B, P, 128)
    float* venc = y + (size_t)B * P * Y_COLS;           // (B, L)
    float* mask_ws = (float*)d_ws;                      // (B, 2, 2048) frag order

    hipLaunchKernelGGL(csg_small_kernel, dim3(B), dim3(256), 0, stream,
                       latent, gu, Kl, Kr, temp, W1, b1, W2, b2, mask_ws, venc);
    hipLaunchKernelGGL(csg_gemm_kernel, dim3(B * (P / ROWS_PER_BLOCK)), dim3(256), 0, stream,
                       x, mask_ws, y);
}